// PathTraversalCPU_14164802142823
// MI455X (gfx1250) — compile-verified
//
#include <hip/hip_runtime.h>

typedef float v4f __attribute__((ext_vector_type(4)));
typedef int   v4i __attribute__((ext_vector_type(4)));

// Pointer types matching the async-LDS builtin signature: V4i in AS1 / AS3.
typedef __attribute__((address_space(1))) v4i* gv4i_p;
typedef __attribute__((address_space(3))) v4i* lv4i_p;

namespace {
constexpr int kH = 224, kW = 224;
constexpr int kB = 8, kC = 64, kP = 8;
constexpr int kL = kH * kW;            // 50176
constexpr int LTILE = 1024;            // l-values per block (4 per thread)
constexpr int NTILE = kL / LTILE;      // 49 (exact)
constexpr int THREADS = 256;           // 8 wave32s
}

#if defined(__has_builtin)
#  if __has_builtin(__builtin_amdgcn_global_load_async_to_lds_b128) && \
      __has_builtin(__builtin_amdgcn_s_wait_asynccnt)
#    define USE_ASYNC_LDS 1
#  else
#    define USE_ASYNC_LDS 0
#  endif
#else
#  define USE_ASYNC_LDS 0
#endif

__global__ __launch_bounds__(THREADS) void path_gather_kernel(
    const float* __restrict__ img, const int* __restrict__ paths,
    float* __restrict__ out)
{
  // 8 KB staging buffer for this tile's (h,w) pairs.
  __shared__ int lds_pairs[LTILE * 2];

  const int tid = threadIdx.x;
  const int blk = blockIdx.x;
  const int lt  = blk % NTILE;
  const int bp  = blk / NTILE;
  const int p   = bp % kP;
  const int b   = bp / kP;
  const int l0  = lt * LTILE;

  // This thread's 4 (h,w) pairs: 8 consecutive ints = 32 bytes.
  const int* gsrc = paths + ((size_t)p * kL + l0) * 2 + (size_t)tid * 8;

#if USE_ASYNC_LDS
  {
    // gfx1250 async copy: global -> LDS, tracked by ASYNCcnt.
    __builtin_amdgcn_global_load_async_to_lds_b128(
        (gv4i_p)(gsrc),     (lv4i_p)(&lds_pairs[tid * 8]),     0, 0);
    __builtin_amdgcn_global_load_async_to_lds_b128(
        (gv4i_p)(gsrc + 4), (lv4i_p)(&lds_pairs[tid * 8 + 4]), 0, 0);
    __builtin_amdgcn_s_wait_asynccnt(0);
  }
#else
  *(v4i*)(&lds_pairs[tid * 8])     = *(const v4i*)(gsrc);
  *(v4i*)(&lds_pairs[tid * 8 + 4]) = *(const v4i*)(gsrc + 4);
#endif
  __syncthreads();

  // Pull pairs back out of LDS (ds_load_b128) and fold into flat indices,
  // reused across all 64 channels from registers.
  v4i q0 = *(const v4i*)(&lds_pairs[tid * 8]);
  v4i q1 = *(const v4i*)(&lds_pairs[tid * 8 + 4]);
  const int i0 = q0.x * kW + q0.y;
  const int i1 = q0.z * kW + q0.w;
  const int i2 = q1.x * kW + q1.y;
  const int i3 = q1.z * kW + q1.w;

  const float* imgb = img + (size_t)b * kC * kL;
  float* outp = out + (((size_t)b * kP + p) * (size_t)kC) * kL + l0 + tid * 4;

  // Per channel: 4 random B32 gather loads (expected L2-resident: img is
  // 102.8 MB vs 192 MB L2) + one contiguous 16B non-temporal store so the
  // 822 MB output stream does not evict img from L2.
#pragma unroll 4
  for (int c = 0; c < kC; ++c) {
    const float* pl = imgb + (size_t)c * kL;
    v4f v;
    v.x = pl[i0];
    v.y = pl[i1];
    v.z = pl[i2];
    v.w = pl[i3];
    __builtin_nontemporal_store(v, (v4f*)outp);
    outp += kL;
  }
}

extern "C" void kernel_launch(void* const* d_in, const int* in_sizes, int n_in,
                              void* d_out, int out_size, void* d_ws, size_t ws_size,
                              hipStream_t stream) {
  const float* img   = (const float*)d_in[0];   // [8, 64, 224, 224] f32
  const int*   paths = (const int*)d_in[1];     // [8, 50176, 2] i32
  float*       out   = (float*)d_out;           // [8, 512, 50176] f32

  dim3 grid(kB * kP * NTILE);                   // 3136 blocks
  path_gather_kernel<<<grid, THREADS, 0, stream>>>(img, paths, out);
}